// SugenoNet_37117107372134
// MI455X (gfx1250) — compile-verified
//
#include <hip/hip_runtime.h>

typedef __attribute__((ext_vector_type(2))) float v2f;
typedef __attribute__((ext_vector_type(8))) float v8f;

#define NUM_RULES   6561                       // 3^8
#define BATCH       8192
#define RT_TILES    ((NUM_RULES + 15) / 16)    // 411 rule tiles of 16
#define CHUNKS      ((RT_TILES + 7) / 8)       // 52 chunks of 8 tiles
#define NWAVES      8                          // waves per workgroup
#define BT          2                          // batch tiles (of 16 rows) per wave

__global__ __launch_bounds__(256) void sugeno_wmma_kernel(
    const float* __restrict__ x,        // [BATCH, 8]
    const float* __restrict__ centers,  // [8, 3]
    const float* __restrict__ sigmas,   // [3]
    const float* __restrict__ cons,     // [NUM_RULES, 9]
    float* __restrict__ out)            // [BATCH]
{
    // B-fragment staging: 8 tiles * 9 fragments (3 cons + 6 E) * 32 lanes * v2f
    __shared__ v2f smem[NWAVES * 9 * 32];      // 18 KB

    const int lane = threadIdx.x & 31;
    const int wave = threadIdx.x >> 5;
    const int half = lane >> 4;      // 0: lanes 0-15, 1: lanes 16-31
    const int col  = lane & 15;      // M (A rows) / N (B cols) index

    const int wb = blockIdx.x * (NWAVES * BT * 16) + wave * (BT * 16); // 32 rows/wave

    // ---------------- per-row setup (A fragments + analytic denominator) ----------
    float inv_sig[3];
    #pragma unroll
    for (int m = 0; m < 3; ++m) inv_sig[m] = 1.0f / sigmas[m];

    v2f  a_xb[BT][3];   // GEMM1 A: XB (16 x 12), K padded from 9
    v2f  a_L [BT][6];   // GEMM2 A: log-memberships L (16 x 24)
    float den[BT];

    #pragma unroll
    for (int bt = 0; bt < BT; ++bt) {
        const int row = wb + bt * 16 + col;

        float xv[8];
        #pragma unroll
        for (int j = 0; j < 8; ++j) xv[j] = x[row * 8 + j];

        float L[24];
        #pragma unroll
        for (int j = 0; j < 8; ++j) {
            #pragma unroll
            for (int m = 0; m < 3; ++m) {
                float z = (xv[j] - centers[j * 3 + m]) * inv_sig[m];
                L[j * 3 + m] = -0.5f * z * z;
            }
        }

        // sum_r s[b,r] factorizes: prod_j (mu_j0 + mu_j1 + mu_j2)
        float d = 1.0f;
        #pragma unroll
        for (int j = 0; j < 8; ++j)
            d *= __expf(L[j * 3 + 0]) + __expf(L[j * 3 + 1]) + __expf(L[j * 3 + 2]);
        den[bt] = d;

        // A(16x4) f32 layout: lane -> M = lane&15 ; VGPR v holds K = 4f + 2*half + v
        #pragma unroll
        for (int f = 0; f < 3; ++f) {
            #pragma unroll
            for (int v = 0; v < 2; ++v) {
                const int k0 = 4 * f + v;       // half == 0
                const int k1 = 4 * f + 2 + v;   // half == 1
                float v0 = (k0 < 8) ? xv[k0] : (k0 == 8 ? 1.0f : 0.0f);
                float v1 = (k1 < 8) ? xv[k1] : (k1 == 8 ? 1.0f : 0.0f);
                a_xb[bt][f][v] = half ? v1 : v0;
            }
        }
        #pragma unroll
        for (int f = 0; f < 6; ++f) {
            #pragma unroll
            for (int v = 0; v < 2; ++v) {
                const int k0 = 4 * f + v;
                const int k1 = 4 * f + 2 + v;
                a_L[bt][f][v] = half ? L[k1] : L[k0];
            }
        }
    }

    const v8f zero = (v8f)0.0f;
    v8f num[BT];
    #pragma unroll
    for (int bt = 0; bt < BT; ++bt) num[bt] = zero;

    // ---------------- sweep rule tiles in workgroup-cooperative chunks ------------
    for (int c = 0; c < CHUNKS; ++c) {
        // ---- build phase: this wave builds B fragments for tile t = c*8 + wave ----
        {
            const int  t      = c * NWAVES + wave;
            const int  rn     = t * 16 + col;            // rule column for this lane
            const bool rvalid = (rn < NUM_RULES);        // covers t >= RT_TILES too

            __builtin_prefetch(cons + (size_t)(c + 1) * NWAVES * 16 * 9, 0, 1);

            // base-3 digits of rn, d0 most significant
            int dg[8];
            {
                int tt = rn;
                #pragma unroll
                for (int j = 7; j >= 0; --j) { int q = tt / 3; dg[j] = tt - q * 3; tt = q; }
            }

            // GEMM1 B: C^T (12 x 16): B[k][n] = cons[n*9 + k], k < 9
            #pragma unroll
            for (int f = 0; f < 3; ++f) {
                v2f bc;
                #pragma unroll
                for (int v = 0; v < 2; ++v) {
                    const int k = 4 * f + 2 * half + v;
                    float val = 0.0f;
                    if (rvalid && k < 9) val = cons[rn * 9 + k];
                    bc[v] = val;
                }
                smem[(wave * 9 + f) * 32 + lane] = bc;
            }

            // GEMM2 B: E (24 x 16): E[(j,m)][n] = (digit_j(rn) == m)
            #pragma unroll
            for (int f = 0; f < 6; ++f) {
                v2f be;
                #pragma unroll
                for (int v = 0; v < 2; ++v) {
                    const int k0 = 4 * f + v;        // half == 0 -> kk = k0
                    const int k1 = 4 * f + 2 + v;    // half == 1 -> kk = k1
                    float e0 = (dg[k0 / 3] == (k0 % 3)) ? 1.0f : 0.0f;
                    float e1 = (dg[k1 / 3] == (k1 % 3)) ? 1.0f : 0.0f;
                    float e  = half ? e1 : e0;
                    be[v] = rvalid ? e : 0.0f;
                }
                smem[(wave * 9 + 3 + f) * 32 + lane] = be;
            }
        }
        __syncthreads();

        // ---- use phase: every wave consumes all 8 staged tiles --------------------
        #pragma unroll 2
        for (int tt = 0; tt < NWAVES; ++tt) {
            v2f bc[3], be[6];
            #pragma unroll
            for (int f = 0; f < 3; ++f) bc[f] = smem[(tt * 9 + f) * 32 + lane];
            #pragma unroll
            for (int f = 0; f < 6; ++f) be[f] = smem[(tt * 9 + 3 + f) * 32 + lane];

            #pragma unroll
            for (int bt = 0; bt < BT; ++bt) {
                // rule_outputs tile = XB * C^T
                v8f ro = zero;
                #pragma unroll
                for (int f = 0; f < 3; ++f)
                    ro = __builtin_amdgcn_wmma_f32_16x16x4_f32(
                             false, a_xb[bt][f], false, bc[f], (short)0, ro, false, false);

                // log-strength tile = L * E
                v8f sl = zero;
                #pragma unroll
                for (int f = 0; f < 6; ++f)
                    sl = __builtin_amdgcn_wmma_f32_16x16x4_f32(
                             false, a_L[bt][f], false, be[f], (short)0, sl, false, false);

                // num += exp(log s) * rule_outputs (invalid cols: ro = 0)
                #pragma unroll
                for (int v = 0; v < 8; ++v)
                    num[bt][v] += __expf(sl[v]) * ro[v];
            }
        }
        __syncthreads();
    }

    // ---------------- reduce over rule columns and write --------------------------
    #pragma unroll
    for (int bt = 0; bt < BT; ++bt) {
        #pragma unroll
        for (int v = 0; v < 8; ++v) {
            float t = num[bt][v];
            t += __shfl_xor(t, 1, 32);
            t += __shfl_xor(t, 2, 32);
            t += __shfl_xor(t, 4, 32);
            t += __shfl_xor(t, 8, 32);
            num[bt][v] = t;          // row (v + 8*half) total, uniform within half
        }

        float outv[8];
        #pragma unroll
        for (int v = 0; v < 8; ++v) {
            float d = __shfl(den[bt], half * 8 + v, 32);
            outv[v] = num[bt][v] / (d + 1e-6f);
        }

        if (col == 0) {  // lane 0 -> rows 0..7, lane 16 -> rows 8..15
            #pragma unroll
            for (int v = 0; v < 8; ++v)
                out[wb + bt * 16 + half * 8 + v] = outv[v];
        }
    }
}

extern "C" void kernel_launch(void* const* d_in, const int* in_sizes, int n_in,
                              void* d_out, int out_size, void* d_ws, size_t ws_size,
                              hipStream_t stream) {
    const float* x       = (const float*)d_in[0];
    const float* centers = (const float*)d_in[1];
    const float* sigmas  = (const float*)d_in[2];
    const float* cons    = (const float*)d_in[3];
    float* out           = (float*)d_out;

    dim3 block(256);                         // 8 waves * 32 rows = 256 rows per WG
    dim3 grid(BATCH / (NWAVES * BT * 16));   // 32 workgroups
    sugeno_wmma_kernel<<<grid, block, 0, stream>>>(x, centers, sigmas, cons, out);
}